// GCNModel_31894427140389
// MI455X (gfx1250) — compile-verified
//
#include <hip/hip_runtime.h>
#include <hip/hip_bf16.h>

#define NN 100000
#define EE 1600000
#define HD 128
#define CC 40

typedef float v2f __attribute__((ext_vector_type(2)));
typedef float v8f __attribute__((ext_vector_type(8)));

// ---------------- degree / normalization ----------------
__global__ void k_deg_init(float* deg, int n) {
    int i = blockIdx.x * blockDim.x + threadIdx.x;
    if (i < n) deg[i] = 1.0f;   // self-loop
}

__global__ void k_deg_acc(float* deg, const int* __restrict__ dst, int e) {
    int i = blockIdx.x * blockDim.x + threadIdx.x;
    if (i < e) unsafeAtomicAdd(&deg[dst[i]], 1.0f);
}

__global__ void k_deg_fin(float* dinv, float* dinv2, int n) {
    int i = blockIdx.x * blockDim.x + threadIdx.x;
    if (i < n) {
        float d = dinv[i];            // holds degree
        dinv[i]  = rsqrtf(d);
        dinv2[i] = 1.0f / d;
    }
}

// ---------------- GEMM: H = X @ W, K=N=128, WMMA f32 16x16x4 ----------------
__global__ __launch_bounds__(256) void k_gemm128_wmma(
    const float* __restrict__ X, const float* __restrict__ W,
    float* __restrict__ Hout, int nrows)
{
    __shared__ float sW[128 * 128];      // 64 KB
    __shared__ float sX[128 * 129];      // 64.5 KB, padded stride kills bank conflicts

    const int tid  = threadIdx.x;
    const int lane = tid & 31;
    const int wv   = tid >> 5;           // wave 0..7
    const int rowBase = blockIdx.x * 128;

    // stage W (contiguous float4 copy)
    {
        const float4* Wv = (const float4*)W;
        float4*       Sv = (float4*)sW;
        #pragma unroll
        for (int i = 0; i < 16; ++i) Sv[tid + i * 256] = Wv[tid + i * 256];
    }
    // stage X tile with zero padding for rows >= nrows
    #pragma unroll
    for (int i = 0; i < 16; ++i) {
        int q  = tid + i * 256;          // 0..4095 float4 slots
        int r  = q >> 5;                 // local row 0..127
        int c4 = (q & 31) << 2;          // col 0..124
        int gr = rowBase + r;
        float4 v = make_float4(0.f, 0.f, 0.f, 0.f);
        if (gr < nrows) v = *(const float4*)(X + (size_t)gr * 128 + c4);
        float* p = &sX[r * 129 + c4];
        p[0] = v.x; p[1] = v.y; p[2] = v.z; p[3] = v.w;
    }
    __syncthreads();

    v8f acc[8] = {};
    const int half = lane >> 4;          // 0: K=k,k+1   1: K=k+2,k+3
    const int l16  = lane & 15;
    const int aRow = wv * 16 + l16;

    for (int k = 0; k < 128; k += 4) {
        const int kk = k + (half << 1);
        v2f a;
        a.x = sX[aRow * 129 + kk];
        a.y = sX[aRow * 129 + kk + 1];
        #pragma unroll
        for (int t = 0; t < 8; ++t) {
            const int col = t * 16 + l16;
            v2f b;
            b.x = sW[kk * 128 + col];
            b.y = sW[(kk + 1) * 128 + col];
            acc[t] = __builtin_amdgcn_wmma_f32_16x16x4_f32(
                false, a, false, b, (short)0, acc[t], false, false);
        }
    }

    #pragma unroll
    for (int t = 0; t < 8; ++t) {
        const int col = t * 16 + l16;
        #pragma unroll
        for (int v = 0; v < 8; ++v) {
            int gr = rowBase + wv * 16 + v + (half << 3);
            if (gr < nrows) Hout[(size_t)gr * 128 + col] = acc[t][v];
        }
    }
}

// ---------------- agg = h*dinv2 + b ----------------
__global__ void k_init_agg(const float* __restrict__ h, const float* __restrict__ dinv2,
                           const float* __restrict__ b, float* __restrict__ agg, int n)
{
    int idx = blockIdx.x * blockDim.x + threadIdx.x;   // n*32 threads
    if (idx >= n * 32) return;
    int node = idx >> 5, q = idx & 31;
    float4 hv = ((const float4*)(h + (size_t)node * 128))[q];
    float4 bv = ((const float4*)b)[q];
    float d2 = dinv2[node];
    float4 r = make_float4(hv.x * d2 + bv.x, hv.y * d2 + bv.y,
                           hv.z * d2 + bv.z, hv.w * d2 + bv.w);
    ((float4*)(agg + (size_t)node * 128))[q] = r;
}

// ---------------- edge scatter: agg[dst] += enorm * h[src] ----------------
__global__ __launch_bounds__(256) void k_scatter(
    const float* __restrict__ h, const int* __restrict__ src, const int* __restrict__ dst,
    const float* __restrict__ dinv, float* __restrict__ agg, int e)
{
    int lane = threadIdx.x & 31;
    int edge = blockIdx.x * 8 + (threadIdx.x >> 5);
    if (edge >= e) return;
    int s = src[edge], d = dst[edge];
    float w = dinv[s] * dinv[d];
    float4 hv = ((const float4*)(h + (size_t)s * 128))[lane];
    float* ap = agg + (size_t)d * 128 + lane * 4;
    unsafeAtomicAdd(ap + 0, w * hv.x);
    unsafeAtomicAdd(ap + 1, w * hv.y);
    unsafeAtomicAdd(ap + 2, w * hv.z);
    unsafeAtomicAdd(ap + 3, w * hv.w);
}

// ---------------- fused LayerNorm + ELU + residual (in-place capable) ----------------
__global__ __launch_bounds__(256) void k_ln_elu_res(
    const float* __restrict__ agg, const float* __restrict__ xres,
    const float* __restrict__ g, const float* __restrict__ be,
    float* __restrict__ out, int useRes, int n)
{
    int lane = threadIdx.x & 31;
    int row  = blockIdx.x * 8 + (threadIdx.x >> 5);
    if (row >= n) return;
    const float4 v = *(const float4*)(agg + (size_t)row * 128 + lane * 4);
    float s  = v.x + v.y + v.z + v.w;
    float sq = v.x * v.x + v.y * v.y + v.z * v.z + v.w * v.w;
    #pragma unroll
    for (int m = 16; m >= 1; m >>= 1) {
        s  += __shfl_xor(s,  m, 32);
        sq += __shfl_xor(sq, m, 32);
    }
    float mu   = s * (1.0f / 128.0f);
    float var  = sq * (1.0f / 128.0f) - mu * mu;
    float rstd = rsqrtf(var + 1e-5f);
    float4 gv = *(const float4*)(g  + lane * 4);
    float4 bv = *(const float4*)(be + lane * 4);
    float4 rv = make_float4(0.f, 0.f, 0.f, 0.f);
    if (useRes) rv = *(const float4*)(xres + (size_t)row * 128 + lane * 4);
    float in[4]  = {v.x, v.y, v.z, v.w};
    float gg[4]  = {gv.x, gv.y, gv.z, gv.w};
    float bb[4]  = {bv.x, bv.y, bv.z, bv.w};
    float rr[4]  = {rv.x, rv.y, rv.z, rv.w};
    float o[4];
    #pragma unroll
    for (int j = 0; j < 4; ++j) {
        float t = (in[j] - mu) * rstd * gg[j] + bb[j];
        t = (t > 0.0f) ? t : expm1f(t);
        o[j] = t + rr[j];
    }
    *(float4*)(out + (size_t)row * 128 + lane * 4) = make_float4(o[0], o[1], o[2], o[3]);
}

// ---------------- classifier: out = X @ Wc + bc (Wc padded to 128x48 in LDS) ----------------
__global__ __launch_bounds__(256) void k_classifier_wmma(
    const float* __restrict__ X, const float* __restrict__ Wc,
    const float* __restrict__ bc, float* __restrict__ out, int nrows)
{
    __shared__ float sW[128 * 48];
    const int tid  = threadIdx.x;
    const int lane = tid & 31;
    const int wv   = tid >> 5;
    const int rowBase = blockIdx.x * 128;

    for (int idx = tid; idx < 128 * 48; idx += 256) {
        int r = idx / 48, c = idx - r * 48;
        sW[idx] = (c < CC) ? Wc[r * CC + c] : 0.0f;
    }
    __syncthreads();

    v8f acc[3] = {};
    const int half = lane >> 4;
    const int l16  = lane & 15;
    const int aGRow = rowBase + wv * 16 + l16;
    const bool aOk = aGRow < nrows;

    for (int k = 0; k < 128; k += 4) {
        const int kk = k + (half << 1);
        v2f a;
        a.x = aOk ? X[(size_t)aGRow * 128 + kk] : 0.0f;
        a.y = aOk ? X[(size_t)aGRow * 128 + kk + 1] : 0.0f;
        #pragma unroll
        for (int t = 0; t < 3; ++t) {
            const int col = t * 16 + l16;
            v2f b;
            b.x = sW[kk * 48 + col];
            b.y = sW[(kk + 1) * 48 + col];
            acc[t] = __builtin_amdgcn_wmma_f32_16x16x4_f32(
                false, a, false, b, (short)0, acc[t], false, false);
        }
    }

    #pragma unroll
    for (int t = 0; t < 3; ++t) {
        const int col = t * 16 + l16;
        if (col >= CC) continue;
        float bias = bc[col];
        #pragma unroll
        for (int v = 0; v < 8; ++v) {
            int gr = rowBase + wv * 16 + v + (half << 3);
            if (gr < nrows) out[(size_t)gr * CC + col] = acc[t][v] + bias;
        }
    }
}

extern "C" void kernel_launch(void* const* d_in, const int* in_sizes, int n_in,
                              void* d_out, int out_size, void* d_ws, size_t ws_size,
                              hipStream_t stream) {
    const float* x  = (const float*)d_in[0];
    const int*   ei = (const int*)d_in[1];
    const float* W0 = (const float*)d_in[2];
    const float* b0 = (const float*)d_in[3];
    const float* g0 = (const float*)d_in[4];
    const float* e0 = (const float*)d_in[5];
    const float* W1 = (const float*)d_in[6];
    const float* b1 = (const float*)d_in[7];
    const float* g1 = (const float*)d_in[8];
    const float* e1 = (const float*)d_in[9];
    const float* W2 = (const float*)d_in[10];
    const float* b2 = (const float*)d_in[11];
    const float* g2 = (const float*)d_in[12];
    const float* e2 = (const float*)d_in[13];
    const float* Wc = (const float*)d_in[14];
    const float* bc = (const float*)d_in[15];
    float* out = (float*)d_out;

    float* ws    = (float*)d_ws;
    float* bufA  = ws;                                // N*128
    float* bufB  = bufA + (size_t)NN * 128;           // N*128
    float* bufH  = bufB + (size_t)NN * 128;           // N*128
    float* dinv  = bufH + (size_t)NN * 128;           // N (holds degree first)
    float* dinv2 = dinv + NN;                         // N

    const int* src = ei;
    const int* dst = ei + EE;

    const int gN   = (NN + 255) / 256;
    const int gE   = (EE + 255) / 256;
    const int gRow = (NN + 127) / 128;     // 782 blocks of 128 rows
    const int gAgg = (NN * 32 + 255) / 256;
    const int gEdg = (EE + 7) / 8;         // 8 edges (waves) per block
    const int gLN  = (NN + 7) / 8;

    // degree / normalization
    k_deg_init<<<gN, 256, 0, stream>>>(dinv, NN);
    k_deg_acc<<<gE, 256, 0, stream>>>(dinv, dst, EE);
    k_deg_fin<<<gN, 256, 0, stream>>>(dinv, dinv2, NN);

    // ---- layer 0: in = x, out = bufA ----
    k_gemm128_wmma<<<gRow, 256, 0, stream>>>(x, W0, bufH, NN);
    k_init_agg<<<gAgg, 256, 0, stream>>>(bufH, dinv2, b0, bufA, NN);
    k_scatter<<<gEdg, 256, 0, stream>>>(bufH, src, dst, dinv, bufA, EE);
    k_ln_elu_res<<<gLN, 256, 0, stream>>>(bufA, nullptr, g0, e0, bufA, 0, NN);

    // ---- layer 1: in = bufA, out = bufB, residual from bufA ----
    k_gemm128_wmma<<<gRow, 256, 0, stream>>>(bufA, W1, bufH, NN);
    k_init_agg<<<gAgg, 256, 0, stream>>>(bufH, dinv2, b1, bufB, NN);
    k_scatter<<<gEdg, 256, 0, stream>>>(bufH, src, dst, dinv, bufB, EE);
    k_ln_elu_res<<<gLN, 256, 0, stream>>>(bufB, bufA, g1, e1, bufB, 1, NN);

    // ---- layer 2: in = bufB, out = bufA, residual from bufB ----
    k_gemm128_wmma<<<gRow, 256, 0, stream>>>(bufB, W2, bufH, NN);
    k_init_agg<<<gAgg, 256, 0, stream>>>(bufH, dinv2, b2, bufA, NN);
    k_scatter<<<gEdg, 256, 0, stream>>>(bufH, src, dst, dinv, bufA, EE);
    k_ln_elu_res<<<gLN, 256, 0, stream>>>(bufA, bufB, g2, e2, bufA, 1, NN);

    // ---- classifier ----
    k_classifier_wmma<<<gRow, 256, 0, stream>>>(bufA, Wc, bc, out, NN);
}